// SAGE_model_44418551775831
// MI455X (gfx1250) — compile-verified
//
#include <hip/hip_runtime.h>
#include <math.h>

// ---------------------------------------------------------------------------
// GraphSAGE forward: 4x SAGEConv(mean)+ReLU, then Linear+sigmoid.
//   N=100000, E=1600000, dims 3->32->64->96->128->4.
// Bottleneck is the edge scatter: minimize index traffic (int32, 1 index pair
// per 4 features), float4 gathers, f32 atomics. GEMMs use V_WMMA_F32_16X16X4
// with weights staged in LDS (zero-padded K) per block.
// ---------------------------------------------------------------------------

#define GN 100000
#define GE 1600000

typedef __attribute__((ext_vector_type(2))) float v2f;
typedef __attribute__((ext_vector_type(8))) float v8f;

// ---------------- zero fill ----------------
__global__ void zero_f32(float* __restrict__ p, long long n) {
    long long i = (long long)blockIdx.x * blockDim.x + threadIdx.x;
    long long stride = (long long)gridDim.x * blockDim.x;
    for (; i < n; i += stride) p[i] = 0.0f;
}

// ---------------- int64 edge_index -> int32 src/dst ----------------
__global__ void cvt_idx_kernel(const long long* __restrict__ ei,
                               int* __restrict__ s32, int* __restrict__ d32,
                               int e_total) {
    int e = blockIdx.x * blockDim.x + threadIdx.x;
    if (e >= e_total) return;
    s32[e] = (int)ei[e];
    d32[e] = (int)ei[(long long)GE + e];
}

// ---------------- degree: deg[dst[e]] += 1 ----------------
__global__ void degree_kernel(const int* __restrict__ dst,
                              float* __restrict__ deg, int e_total) {
    int e = blockIdx.x * blockDim.x + threadIdx.x;
    if (e >= e_total) return;
    __hip_atomic_fetch_add(&deg[dst[e]], 1.0f, __ATOMIC_RELAXED,
                           __HIP_MEMORY_SCOPE_AGENT);
}

// ---------------- scalar scatter (din=3) ----------------
__global__ void scatter_add_scalar(const float* __restrict__ h,
                                   const int* __restrict__ src,
                                   const int* __restrict__ dst,
                                   float* __restrict__ agg,
                                   int din, long long total) {
    long long i = (long long)blockIdx.x * blockDim.x + threadIdx.x;
    long long stride = (long long)gridDim.x * blockDim.x;
    for (; i < total; i += stride) {
        long long e = i / din;
        int f = (int)(i - e * din);
        int s = src[e], d = dst[e];
        float v = h[(long long)s * din + f];
        __hip_atomic_fetch_add(&agg[(long long)d * din + f], v,
                               __ATOMIC_RELAXED, __HIP_MEMORY_SCOPE_AGENT);
    }
}

// ---------------- vec4 scatter: 1 index pair per 4 features ----------------
template <int DIN>
__global__ void scatter_add_vec4(const float* __restrict__ h,
                                 const int* __restrict__ src,
                                 const int* __restrict__ dst,
                                 float* __restrict__ agg, long long total) {
    constexpr int C4 = DIN / 4;
    long long i = (long long)blockIdx.x * blockDim.x + threadIdx.x;
    long long stride = (long long)gridDim.x * blockDim.x;
    for (; i < total; i += stride) {
        long long e = i / C4;
        int c = (int)(i - e * C4);
        int s = src[e], d = dst[e];
        float4 v = reinterpret_cast<const float4*>(h)[(long long)s * C4 + c];
        float* p = agg + (long long)d * DIN + (long long)c * 4;
        __hip_atomic_fetch_add(p + 0, v.x, __ATOMIC_RELAXED, __HIP_MEMORY_SCOPE_AGENT);
        __hip_atomic_fetch_add(p + 1, v.y, __ATOMIC_RELAXED, __HIP_MEMORY_SCOPE_AGENT);
        __hip_atomic_fetch_add(p + 2, v.z, __ATOMIC_RELAXED, __HIP_MEMORY_SCOPE_AGENT);
        __hip_atomic_fetch_add(p + 3, v.w, __ATOMIC_RELAXED, __HIP_MEMORY_SCOPE_AGENT);
    }
}

// ---------------- SAGE layer GEMM via WMMA f32 16x16x4, LDS-staged B ------
// out[m,:] = relu( (agg[m,:]/max(deg[m],1)) @ Wl + h[m,:] @ Wr + bl )
// Block = 4 waves; each wave owns a 16-row tile. Per pass, the active weight
// matrix (zero-padded to DINP rows) is staged in LDS, then K-stepped WMMA.
// A frag (16x4 f32): lane L -> M=L%16, K=(L/16)*2+{0,1}.
// B frag (4x16 f32): lane L -> N=L%16, K=(L/16)*2+{0,1}.
// D (16x16 f32):     lane L, VGPR v -> M=v+8*(L/16), N=L%16.
template <int DIN, int NT>
__global__ void sage_gemm_wmma(const float* __restrict__ agg,
                               const float* __restrict__ deg,
                               const float* __restrict__ h,
                               const float* __restrict__ Wl,
                               const float* __restrict__ bl,
                               const float* __restrict__ Wr,
                               float* __restrict__ out, int n) {
    constexpr int DOUT = NT * 16;
    constexpr int DINP = (DIN + 3) & ~3;        // K padded to multiple of 4
    constexpr int NVEC = DINP * DOUT / 4;       // float4 elements in LDS tile
    extern __shared__ float sW[];               // DINP * DOUT floats (<=48KB)

    const int tid  = threadIdx.x;
    const int lane = tid & 31;
    const int wave = tid >> 5;
    const int tile = blockIdx.x * 4 + wave;
    const int m0   = tile * 16;
    const bool valid = (m0 < n);                // wave-uniform

    const int mrow  = m0 + (lane & 15);
    const int khalf = (lane >> 4) * 2;          // 0 or 2
    const int ncol  = lane & 15;

    float inv = 1.0f;
    if (valid) inv = 1.0f / fmaxf(deg[mrow], 1.0f);

    v8f acc[NT] = {};

    #pragma unroll
    for (int pass = 0; pass < 2; ++pass) {
        const float* __restrict__ W = pass ? Wr : Wl;
        __syncthreads();                        // previous pass done with sW
        #pragma unroll 2
        for (int i = tid; i < NVEC; i += 128) { // stage + zero-pad weights
            int row = i / (DOUT / 4);
            float4 v = make_float4(0.f, 0.f, 0.f, 0.f);
            if (row < DIN)
                v = reinterpret_cast<const float4*>(W)[i];
            reinterpret_cast<float4*>(sW)[i] = v;
        }
        __syncthreads();

        if (valid) {
            const float* __restrict__ A = pass ? h : agg;
            const float scale = pass ? 1.0f : inv;
            const long long arow = (long long)mrow * DIN;
            #pragma unroll
            for (int k = 0; k < DINP; k += 4) {
                const int k0 = k + khalf;
                const int k1 = k0 + 1;
                v2f a;
                if constexpr (DIN % 4 == 0) {   // contiguous, 8B-aligned pair
                    a = *reinterpret_cast<const v2f*>(A + arow + k0);
                } else {                        // tail layer (din=3)
                    a.x = (k0 < DIN) ? A[arow + k0] : 0.0f;
                    a.y = (k1 < DIN) ? A[arow + k1] : 0.0f;
                }
                a.x *= scale;
                a.y *= scale;
                #pragma unroll
                for (int t = 0; t < NT; ++t) {  // B from LDS (padded: no guards)
                    v2f b;
                    b.x = sW[k0 * DOUT + t * 16 + ncol];
                    b.y = sW[k1 * DOUT + t * 16 + ncol];
                    acc[t] = __builtin_amdgcn_wmma_f32_16x16x4_f32(
                        false, a, false, b, (short)0, acc[t], false, false);
                }
            }
        }
    }

    if (!valid) return;
    // epilogue: bias + relu + store (lanes coalesced along ncol)
    const int rbase = m0 + 8 * (lane >> 4);
    #pragma unroll
    for (int t = 0; t < NT; ++t) {
        const float bias = bl[t * 16 + ncol];
        #pragma unroll
        for (int v = 0; v < 8; ++v) {
            float val = fmaxf(acc[t][v] + bias, 0.0f);
            out[(long long)(rbase + v) * DOUT + t * 16 + ncol] = val;
        }
    }
}

// ---------------- final: sigmoid(h @ Wf + bf), DOUT=4 ----------------
__global__ void final_kernel(const float* __restrict__ h,
                             const float* __restrict__ Wf,
                             const float* __restrict__ bf,
                             float* __restrict__ out, int n) {
    int idx = blockIdx.x * blockDim.x + threadIdx.x;
    if (idx >= n * 4) return;
    int i = idx >> 2;
    int c = idx & 3;
    float s = bf[c];
    const float4* h4 = reinterpret_cast<const float4*>(h + (long long)i * 128);
    #pragma unroll 8
    for (int k4 = 0; k4 < 32; ++k4) {
        float4 hv = h4[k4];
        s += hv.x * Wf[(k4 * 4 + 0) * 4 + c];
        s += hv.y * Wf[(k4 * 4 + 1) * 4 + c];
        s += hv.z * Wf[(k4 * 4 + 2) * 4 + c];
        s += hv.w * Wf[(k4 * 4 + 3) * 4 + c];
    }
    out[idx] = 1.0f / (1.0f + expf(-s));
}

// ---------------------------------------------------------------------------
extern "C" void kernel_launch(void* const* d_in, const int* in_sizes, int n_in,
                              void* d_out, int out_size, void* d_ws, size_t ws_size,
                              hipStream_t stream) {
    const float*     x  = (const float*)d_in[0];
    const long long* ei = (const long long*)d_in[1];   // int64 edge_index [2,E]
    const float* Wl0 = (const float*)d_in[2];
    const float* bl0 = (const float*)d_in[3];
    const float* Wr0 = (const float*)d_in[4];
    const float* Wl1 = (const float*)d_in[5];
    const float* bl1 = (const float*)d_in[6];
    const float* Wr1 = (const float*)d_in[7];
    const float* Wl2 = (const float*)d_in[8];
    const float* bl2 = (const float*)d_in[9];
    const float* Wr2 = (const float*)d_in[10];
    const float* Wl3 = (const float*)d_in[11];
    const float* bl3 = (const float*)d_in[12];
    const float* Wr3 = (const float*)d_in[13];
    const float* Wf  = (const float*)d_in[14];
    const float* bf  = (const float*)d_in[15];
    float* out = (float*)d_out;

    // workspace layout
    float* ws   = (float*)d_ws;
    float* deg  = ws;                               // N
    float* agg  = deg  + GN;                        // N*96 max
    float* bufA = agg  + (long long)GN * 96;        // N*96  (h1[32], h3[96])
    float* bufB = bufA + (long long)GN * 96;        // N*128 (h2[64], h4[128])
    int*   s32  = (int*)(bufB + (long long)GN * 128); // E
    int*   d32  = s32 + GE;                         // E

    const int T = 256;
    const int tiles = GN / 16;                      // 100000 % 16 == 0
    const int gemm_grid = (tiles + 3) / 4;          // 4 waves (128 thr)/block

    // indices -> int32, degree (shared by all layers)
    cvt_idx_kernel<<<(GE + T - 1) / T, T, 0, stream>>>(ei, s32, d32, GE);
    zero_f32<<<(GN + T - 1) / T, T, 0, stream>>>(deg, GN);
    degree_kernel<<<(GE + T - 1) / T, T, 0, stream>>>(d32, deg, GE);

    // ---- layer 0: 3 -> 32
    {
        long long tot = (long long)GE * 3;
        zero_f32<<<(GN * 3 + T - 1) / T, T, 0, stream>>>(agg, (long long)GN * 3);
        scatter_add_scalar<<<(int)((tot + T - 1) / T), T, 0, stream>>>(x, s32, d32, agg, 3, tot);
        sage_gemm_wmma<3, 2><<<gemm_grid, 128, 4 * 32 * 4, stream>>>(
            agg, deg, x, Wl0, bl0, Wr0, bufA, GN);
    }
    // ---- layer 1: 32 -> 64
    {
        long long tot = (long long)GE * (32 / 4);
        zero_f32<<<(GN * 32 + T - 1) / T, T, 0, stream>>>(agg, (long long)GN * 32);
        scatter_add_vec4<32><<<(int)((tot + T - 1) / T), T, 0, stream>>>(bufA, s32, d32, agg, tot);
        sage_gemm_wmma<32, 4><<<gemm_grid, 128, 32 * 64 * 4, stream>>>(
            agg, deg, bufA, Wl1, bl1, Wr1, bufB, GN);
    }
    // ---- layer 2: 64 -> 96
    {
        long long tot = (long long)GE * (64 / 4);
        zero_f32<<<(GN * 64 + T - 1) / T, T, 0, stream>>>(agg, (long long)GN * 64);
        scatter_add_vec4<64><<<(int)((tot + T - 1) / T), T, 0, stream>>>(bufB, s32, d32, agg, tot);
        sage_gemm_wmma<64, 6><<<gemm_grid, 128, 64 * 96 * 4, stream>>>(
            agg, deg, bufB, Wl2, bl2, Wr2, bufA, GN);
    }
    // ---- layer 3: 96 -> 128
    {
        long long tot = (long long)GE * (96 / 4);
        zero_f32<<<(GN * 96 + T - 1) / T, T, 0, stream>>>(agg, (long long)GN * 96);
        scatter_add_vec4<96><<<(int)((tot + T - 1) / T), T, 0, stream>>>(bufA, s32, d32, agg, tot);
        sage_gemm_wmma<96, 8><<<gemm_grid, 128, 96 * 128 * 4, stream>>>(
            agg, deg, bufA, Wl3, bl3, Wr3, bufB, GN);
    }
    // ---- final: 128 -> 4, sigmoid
    final_kernel<<<(GN * 4 + T - 1) / T, T, 0, stream>>>(bufB, Wf, bf, out, GN);
}